// GCNNLayer_3676492005960
// MI455X (gfx1250) — compile-verified
//
#include <hip/hip_runtime.h>

typedef __attribute__((ext_vector_type(2))) float v2f;
typedef __attribute__((ext_vector_type(8))) float v8f;
typedef int v4i __attribute__((vector_size(16)));   // matches builtin's vector pointee type

constexpr int Nn = 4096;   // tokens
constexpr int Dd = 1024;   // hidden
constexpr int Rr = 92;     // relations
constexpr int Tt = 32;     // tags
constexpr int Ee = 8192;   // edges

constexpr int KC   = 256;      // K staging chunk (floats)
constexpr int XP   = KC + 4;   // LDS pitch: 260 -> per-row bank offset 4, conflict-free, 16B aligned
constexpr int NCH  = Dd / KC;  // 4 K-chunks
constexpr int MAXC = 640;      // >= E/16 + R = 604

// ---- CDNA5 async global->LDS copy support (probe-gated) ----
#if defined(__has_builtin)
#if __has_builtin(__builtin_amdgcn_global_load_async_to_lds_b128)
#define HAVE_ASYNC_LDS 1
#endif
#endif
#ifndef HAVE_ASYNC_LDS
#define HAVE_ASYNC_LDS 0
#endif

#if HAVE_ASYNC_LDS
__device__ __forceinline__ void wait_async0() {
#if __has_builtin(__builtin_amdgcn_s_wait_asynccnt)
    __builtin_amdgcn_s_wait_asynccnt(0);
#else
    asm volatile("s_wait_asynccnt 0x0" ::: "memory");
#endif
}
#endif

// ---------------- edge preprocessing ----------------

__global__ void k_zero(int* counts) {
    if (threadIdx.x < Rr) counts[threadIdx.x] = 0;
}

__global__ void k_hist(const int* __restrict__ edge_rel, int* __restrict__ counts) {
    int e = blockIdx.x * blockDim.x + threadIdx.x;
    if (e < Ee) atomicAdd(&counts[edge_rel[e]], 1);
}

__global__ void k_scan_build(const int* __restrict__ counts, int* __restrict__ offsets,
                             int* __restrict__ cursor, int* __restrict__ n_chunks,
                             int* __restrict__ crel, int* __restrict__ cbeg,
                             int* __restrict__ ccnt) {
    if (threadIdx.x != 0) return;
    int acc = 0;
    for (int r = 0; r < Rr; ++r) { offsets[r] = acc; cursor[r] = acc; acc += counts[r]; }
    offsets[Rr] = acc;
    int nc = 0;
    for (int r = 0; r < Rr; ++r) {
        int beg = offsets[r], end = offsets[r + 1];
        for (int s = beg; s < end; s += 16) {
            crel[nc] = r; cbeg[nc] = s; ccnt[nc] = min(16, end - s); ++nc;
        }
    }
    *n_chunks = nc;
}

__global__ void k_scatter(const int* __restrict__ edge_tok, const int* __restrict__ edge_rel,
                          int* __restrict__ cursor, int* __restrict__ stok) {
    int e = blockIdx.x * blockDim.x + threadIdx.x;
    if (e < Ee) {
        int pos = atomicAdd(&cursor[edge_rel[e]], 1);
        stok[pos] = edge_tok[e];
    }
}

// ---------------- WMMA f32 16x16x4 GEMM over 16-row chunks ----------------
// GATHER=false: rows = identity (self GEMM), plain store of x@W_self + b_self.
// GATHER=true : rows gathered via sorted tokens, result atomically added
//               (x_tok @ W_stack[rel] + B_stack[rel]) into acc.
// Block: 128 threads = 4 waves; wave w owns 16-col tile (blockIdx.x*4+w)*16.
// A tile (16 x KC) double-buffered in LDS. Next chunk's copy is issued before
// the WMMA loop of the current chunk (async-to-LDS if available, else a
// register-staged b128 clause), so HBM gather latency overlaps the WMMAs.

template <bool GATHER>
__global__ __launch_bounds__(128)
void gemm16_kernel(const float* __restrict__ x,
                   const float* __restrict__ Wall,
                   const float* __restrict__ Ball,
                   float* __restrict__ acc,
                   const int* __restrict__ sorted_tok,
                   const int* __restrict__ chunk_rel,
                   const int* __restrict__ chunk_beg,
                   const int* __restrict__ chunk_cnt,
                   const int* __restrict__ n_chunks) {
    __shared__ float xt[2][16 * XP];   // ~33 KB double-buffered A tile
    __shared__ int rsrc[16];
    __shared__ int meta[2]; // cnt, rel

    const int tid = threadIdx.x;
    const int cid = blockIdx.y;
    if constexpr (GATHER) {
        if (cid >= *n_chunks) return;
    }

    if (tid < 16) {
        int row, cnt, rel;
        if constexpr (GATHER) {
            rel = chunk_rel[cid];
            int beg = chunk_beg[cid];
            cnt = chunk_cnt[cid];
            row = (tid < cnt) ? sorted_tok[beg + tid] : 0;
        } else {
            rel = 0; cnt = 16;
            row = cid * 16 + tid;
        }
        rsrc[tid] = row;
        if (tid == 0) { meta[0] = cnt; meta[1] = rel; }
    }
    __syncthreads();

    const int cnt = meta[0];
    const int rel = meta[1];
    const float* __restrict__ W    = Wall + (GATHER ? (size_t)rel * Dd * Dd : 0);
    const float* __restrict__ bias = Ball + (GATHER ? (size_t)rel * Dd : 0);

    const int lane = tid & 31;
    const int wid  = tid >> 5;
    const int n16  = lane & 15;         // B col within tile == A row (M) for this lane
    const int hi   = lane >> 4;         // K sub-pair select (ISA A/B fragment layout)
    const int col  = (blockIdx.x * 4 + wid) * 16 + n16;
    const float bv = bias[col];
    const float* __restrict__ Wcol = W + col;   // element [k][col] at Wcol[k*Dd]

    // Per-thread staging plan: thread covers rows m = (tid>>6) + 2u, u=0..7,
    // contiguous float4 at column (tid&63)*4 of each KC chunk.
    // 32-bit element offsets from the uniform x base (saddr+voffset form).
    const int mBase = tid >> 6;
    const int q4    = (tid & 63) * 4;
    int soff[8];
    int dsto[8];
#pragma unroll
    for (int u = 0; u < 8; ++u) {
        const int m = mBase + 2 * u;
        soff[u] = rsrc[m] * Dd + q4;
        dsto[u] = m * XP + q4;
    }

    v8f c = {};

#if HAVE_ASYNC_LDS
    // -------- async-to-LDS pipeline: no VGPR staging at all --------
    auto stage_async = [&](int k0, int buf) {
#pragma unroll
        for (int u = 0; u < 8; ++u) {
            __builtin_amdgcn_global_load_async_to_lds_b128(
                (__attribute__((address_space(1))) v4i*)(uintptr_t)(x + soff[u] + k0),
                (__attribute__((address_space(3))) v4i*)(uint32_t)(uintptr_t)(&xt[buf][dsto[u]]),
                0, 0);
        }
    };
    stage_async(0, 0);
    wait_async0();
    __syncthreads();
#else
    // -------- register-staged fallback pipeline --------
    float4 t[8];
#pragma unroll
    for (int u = 0; u < 8; ++u) t[u] = *(const float4*)(x + soff[u]);
#endif

#pragma unroll
    for (int ch = 0; ch < NCH; ++ch) {
        const int k0  = ch * KC;
        const int buf = ch & 1;

#if HAVE_ASYNC_LDS
        // issue next chunk's async copies into the other buffer
        if (ch + 1 < NCH) stage_async(k0 + KC, buf ^ 1);
#else
        // commit staged registers to this LDS buffer
#pragma unroll
        for (int u = 0; u < 8; ++u) *(float4*)(&xt[buf][dsto[u]]) = t[u];
        __syncthreads();
        // issue next chunk's gather loads before compute (latency overlap)
        if (ch + 1 < NCH) {
#pragma unroll
            for (int u = 0; u < 8; ++u) t[u] = *(const float4*)(x + soff[u] + k0 + KC);
        }
#endif

#pragma unroll 8
        for (int kk = 0; kk < KC; kk += 4) {
            const int ka = kk + 2 * hi;
            // A fragment: lanes 0-15 -> K = kk,kk+1 ; lanes 16-31 -> K = kk+2,kk+3
            v2f a = *(const v2f*)(&xt[buf][n16 * XP + ka]);
            // B fragment mirrors A over N
            v2f b;
            b.x = Wcol[(size_t)(k0 + ka) * Dd];
            b.y = Wcol[(size_t)(k0 + ka + 1) * Dd];
            c = __builtin_amdgcn_wmma_f32_16x16x4_f32(
                    false, a, false, b, (short)0, c, false, false);
        }

#if HAVE_ASYNC_LDS
        wait_async0();     // next buffer's data landed
        __syncthreads();
#else
        __syncthreads();   // buffer may be reused 2 chunks later
#endif
    }

#pragma unroll
    for (int j = 0; j < 8; ++j) {
        const int mo = j + 8 * hi;      // D layout: VGPR j -> M=j (lanes 0-15) / j+8 (16-31)
        const float val = c[j] + bv;
        if constexpr (GATHER) {
            if (mo < cnt) atomicAdd(acc + (size_t)rsrc[mo] * Dd + col, val);
        } else {
            acc[(size_t)rsrc[mo] * Dd + col] = val;
        }
    }
}

// ---------------- relu + hidden2tag + log_softmax ----------------
// One wave per row: lane = tag. W_tag reads are fully coalesced (32 lanes x 4B).

__global__ __launch_bounds__(256)
void tag_kernel(float* __restrict__ acc,             // [N,D] pre-relu, relu'd in place -> h
                const float* __restrict__ Wtag,      // [D,T]
                const float* __restrict__ btag,      // [T]
                float* __restrict__ tag_out) {       // [N,T]
    __shared__ float hrow[8][Dd];
    const int tid = threadIdx.x, lane = tid & 31, wid = tid >> 5;
    const int row = blockIdx.x * 8 + wid;

    float* hp = acc + (size_t)row * Dd;
#pragma unroll
    for (int j = 0; j < Dd / 32; ++j) {
        float v = hp[j * 32 + lane];
        v = v > 0.0f ? v : 0.0f;
        hp[j * 32 + lane] = v;           // final h output
        hrow[wid][j * 32 + lane] = v;
    }
    __syncthreads();

    float t = btag[lane];
#pragma unroll 8
    for (int k = 0; k < Dd; ++k)
        t += hrow[wid][k] * Wtag[(size_t)k * Tt + lane];

    float m = t;
#pragma unroll
    for (int off = 16; off > 0; off >>= 1) m = fmaxf(m, __shfl_xor(m, off, 32));
    float s = __expf(t - m);
#pragma unroll
    for (int off = 16; off > 0; off >>= 1) s += __shfl_xor(s, off, 32);
    tag_out[(size_t)row * Tt + lane] = t - m - __logf(s);
}

// ---------------- launcher ----------------

extern "C" void kernel_launch(void* const* d_in, const int* in_sizes, int n_in,
                              void* d_out, int out_size, void* d_ws, size_t ws_size,
                              hipStream_t stream) {
    (void)in_sizes; (void)n_in; (void)out_size; (void)ws_size;
    const float* x       = (const float*)d_in[0];
    const float* W_self  = (const float*)d_in[1];
    const float* b_self  = (const float*)d_in[2];
    const float* W_stack = (const float*)d_in[3];
    const float* B_stack = (const float*)d_in[4];
    const float* W_tag   = (const float*)d_in[5];
    const float* b_tag   = (const float*)d_in[6];
    const int* edge_tok  = (const int*)d_in[7];
    const int* edge_rel  = (const int*)d_in[8];

    float* out     = (float*)d_out;
    float* tag_out = out;                       // [N,T]
    float* acc     = out + (size_t)Nn * Tt;     // [N,D] accumulator -> h (fully overwritten)

    int* ws      = (int*)d_ws;
    int* counts  = ws;                 // R
    int* offsets = ws + 92;            // R+1
    int* cursor  = ws + 185;           // R
    int* nchunks = ws + 277;           // 1
    int* crel    = ws + 278;           // MAXC
    int* cbeg    = ws + 278 + MAXC;    // MAXC
    int* ccnt    = ws + 278 + 2 * MAXC;// MAXC
    int* stok    = ws + 278 + 3 * MAXC;// E

    k_zero<<<1, 128, 0, stream>>>(counts);
    k_hist<<<Ee / 256, 256, 0, stream>>>(edge_rel, counts);
    k_scan_build<<<1, 32, 0, stream>>>(counts, offsets, cursor, nchunks, crel, cbeg, ccnt);
    k_scatter<<<Ee / 256, 256, 0, stream>>>(edge_tok, edge_rel, cursor, stok);

    // self GEMM: acc = x @ W_self + b_self
    dim3 gs(Dd / 64, Nn / 16);   // 16 col-groups x 256 row-chunks
    gemm16_kernel<false><<<gs, 128, 0, stream>>>(x, W_self, b_self, acc,
                                                 stok, crel, cbeg, ccnt, nchunks);

    // edge GEMM: acc[tok] += x[tok] @ W_stack[rel] + B_stack[rel]  (per edge)
    dim3 ge(Dd / 64, MAXC);
    gemm16_kernel<true><<<ge, 128, 0, stream>>>(x, W_stack, B_stack, acc,
                                                stok, crel, cbeg, ccnt, nchunks);

    // relu + tag scores + log_softmax
    tag_kernel<<<Nn / 8, 256, 0, stream>>>(acc, W_tag, b_tag, tag_out);
}